// Network_21062519620339
// MI455X (gfx1250) — compile-verified
//
#include <hip/hip_runtime.h>

typedef __attribute__((ext_vector_type(2))) float v2f;
typedef __attribute__((ext_vector_type(8))) float v8f;

#define F_IN  128
#define F_OUT 64

// ---------- degree / normalization ----------
__global__ void k_init_deg(float* __restrict__ deg, int n) {
    int i = blockIdx.x * blockDim.x + threadIdx.x;
    if (i < n) deg[i] = 1.0f;               // self-loop weight 1
}

__global__ void k_accum_deg(const int* __restrict__ col, const float* __restrict__ ew,
                            float* __restrict__ deg, int E) {
    int e = blockIdx.x * blockDim.x + threadIdx.x;
    if (e < E) atomicAdd(&deg[col[e]], ew[e]);
}

__global__ void k_deg_to_dis(float* __restrict__ deg, int n) {
    int i = blockIdx.x * blockDim.x + threadIdx.x;
    if (i < n) {
        float d = deg[i];
        deg[i] = (d > 0.0f) ? rsqrtf(d) : 0.0f;   // in-place deg -> dis
    }
}

// ---------- XW = X @ W via V_WMMA_F32_16X16X4_F32 ----------
// One wave computes a 16-row x 64-col output tile (4 accumulators of 16x16),
// looping K=128 in steps of 4. W is staged in LDS (32 KB).
__global__ __launch_bounds__(256) void k_gemm_xw(const float* __restrict__ X,
                                                 const float* __restrict__ Wm,
                                                 float* __restrict__ XW, int n) {
    __shared__ float sW[F_IN * F_OUT];          // 32 KB
    {
        const float4* W4  = (const float4*)Wm;
        float4*       sW4 = (float4*)sW;
        for (int idx = threadIdx.x; idx < (F_IN * F_OUT) / 4; idx += blockDim.x)
            sW4[idx] = W4[idx];
    }
    __syncthreads();

    const int wave = threadIdx.x >> 5;
    const int lane = threadIdx.x & 31;
    const int tile = blockIdx.x * (blockDim.x >> 5) + wave;
    const int ntiles = (n + 15) >> 4;
    if (tile >= ntiles) return;                 // wave-uniform

    const int half = lane >> 4;                 // 0: lanes 0-15, 1: lanes 16-31
    const int lo   = lane & 15;

    // A-matrix row for loads (clamped for the (non-existent here) tail tile)
    int row  = tile * 16 + lo;
    int rowc = (row < n) ? row : (n - 1);
    const float* xr = X + (size_t)rowc * F_IN;

    v8f acc0 = {}, acc1 = {}, acc2 = {}, acc3 = {};

#pragma unroll 8
    for (int k0 = 0; k0 < F_IN; k0 += 4) {
        const int ka = k0 + 2 * half;           // v0 holds K=ka, v1 holds K=ka+1
        v2f a;
        a.x = xr[ka];
        a.y = xr[ka + 1];

        const float* w0 = sW + (size_t)ka * F_OUT;        // row K=ka
        const float* w1 = sW + (size_t)(ka + 1) * F_OUT;  // row K=ka+1

        v2f b;
        b.x = w0[lo +  0]; b.y = w1[lo +  0];
        acc0 = __builtin_amdgcn_wmma_f32_16x16x4_f32(false, a, false, b, (short)0, acc0, false, false);
        b.x = w0[lo + 16]; b.y = w1[lo + 16];
        acc1 = __builtin_amdgcn_wmma_f32_16x16x4_f32(false, a, false, b, (short)0, acc1, false, false);
        b.x = w0[lo + 32]; b.y = w1[lo + 32];
        acc2 = __builtin_amdgcn_wmma_f32_16x16x4_f32(false, a, false, b, (short)0, acc2, false, false);
        b.x = w0[lo + 48]; b.y = w1[lo + 48];
        acc3 = __builtin_amdgcn_wmma_f32_16x16x4_f32(false, a, false, b, (short)0, acc3, false, false);
    }

    // C/D layout: VGPR j -> M = j + 8*half, N = lo (per 16-col tile)
    const int mbase = tile * 16 + 8 * half;
#pragma unroll
    for (int j = 0; j < 8; ++j) {
        int m = mbase + j;
        if (m < n) {
            float* o = XW + (size_t)m * F_OUT + lo;
            o[0]  = acc0[j];
            o[16] = acc1[j];
            o[32] = acc2[j];
            o[48] = acc3[j];
        }
    }
}

// ---------- self-loop term: out[i] = dis[i]^2 * XW[i]  (initializes d_out) ----------
__global__ void k_selfloop(const float* __restrict__ xw, const float* __restrict__ dis,
                           float* __restrict__ out, int n) {
    size_t g = (size_t)blockIdx.x * blockDim.x + threadIdx.x;
    if (g >= (size_t)n * F_OUT) return;
    int node = (int)(g >> 6);
    float d = dis[node];
    out[g] = d * d * xw[g];
}

// ---------- edge scatter: 32 lanes per edge, 2 features per lane ----------
__global__ void k_edge_scatter(const int* __restrict__ row, const int* __restrict__ col,
                               const float* __restrict__ ew, const float* __restrict__ dis,
                               const float* __restrict__ xw, float* __restrict__ out, int E) {
    size_t g = (size_t)blockIdx.x * blockDim.x + threadIdx.x;
    size_t e = g >> 5;
    int lane = threadIdx.x & 31;
    if (e >= (size_t)E) return;
    int s = row[e], t = col[e];
    float norm = dis[s] * ew[e] * dis[t];
    const float2* src = (const float2*)(xw + (size_t)s * F_OUT);
    float2 v = src[lane];                        // coalesced 256B gather per edge
    float* dst = out + (size_t)t * F_OUT + 2 * lane;
    atomicAdd(dst,     norm * v.x);              // L2-resident f32 atomics
    atomicAdd(dst + 1, norm * v.y);
}

// ---------- bias + ReLU ----------
__global__ void k_bias_relu(float* __restrict__ out, const float* __restrict__ b, int n) {
    size_t g = (size_t)blockIdx.x * blockDim.x + threadIdx.x;
    if (g >= (size_t)n * F_OUT) return;
    float v = out[g] + b[g & (F_OUT - 1)];
    out[g] = (v > 0.0f) ? v : 0.0f;
}

extern "C" void kernel_launch(void* const* d_in, const int* in_sizes, int n_in,
                              void* d_out, int out_size, void* d_ws, size_t ws_size,
                              hipStream_t stream) {
    const float* X  = (const float*)d_in[0];   // [N, 128]
    const int*   ei = (const int*)  d_in[1];   // [2, E]
    const float* ew = (const float*)d_in[2];   // [E]
    const float* Wm = (const float*)d_in[3];   // [128, 64]
    const float* b  = (const float*)d_in[4];   // [64]

    const int N = in_sizes[0] / F_IN;
    const int E = in_sizes[2];
    const int* row = ei;          // sources
    const int* col = ei + E;      // targets

    float* xw  = (float*)d_ws;                 // N*64 floats (25.6 MB)
    float* deg = xw + (size_t)N * F_OUT;       // N floats; becomes dis in-place
    float* out = (float*)d_out;

    const int T = 256;
    k_init_deg  <<<(N + T - 1) / T, T, 0, stream>>>(deg, N);
    k_accum_deg <<<(E + T - 1) / T, T, 0, stream>>>(col, ew, deg, E);
    k_deg_to_dis<<<(N + T - 1) / T, T, 0, stream>>>(deg, N);

    const int tiles = (N + 15) / 16;           // 16-row tiles, 1 wave each, 8 waves/block
    k_gemm_xw<<<(tiles + 7) / 8, T, 0, stream>>>(X, Wm, xw, N);

    const size_t tot = (size_t)N * F_OUT;
    k_selfloop<<<(unsigned)((tot + T - 1) / T), T, 0, stream>>>(xw, deg, out, N);

    const size_t ethreads = (size_t)E * 32;
    k_edge_scatter<<<(unsigned)((ethreads + T - 1) / T), T, 0, stream>>>(row, col, ew, deg, xw, out, E);

    k_bias_relu<<<(unsigned)((tot + T - 1) / T), T, 0, stream>>>(out, b, N);
}